// DEC_20873541059056
// MI455X (gfx1250) — compile-verified
//
#include <hip/hip_runtime.h>

typedef __attribute__((ext_vector_type(2))) float v2f;
typedef __attribute__((ext_vector_type(8))) float v8f;

#define TILE_M 64
#define LDA 132  // 128 + 4 floats pad -> conflict-free WMMA A-fragment ds loads

// Pre-kernel: csq[k] = sum_d centers[k][d]^2  (256 values into d_ws)
__global__ void __launch_bounds__(256) dec_csq_kernel(const float* __restrict__ centers,
                                                      float* __restrict__ csq) {
  int k = threadIdx.x;
  const float4* p = (const float4*)(centers + (size_t)k * 128);
  float s = 0.f;
#pragma unroll
  for (int i = 0; i < 32; ++i) {
    float4 v = p[i];
    s += v.x * v.x + v.y * v.y + v.z * v.z + v.w * v.w;
  }
  csq[k] = s;
}

__global__ void __launch_bounds__(256) dec_main_kernel(const float* __restrict__ batch,
                                                       const float* __restrict__ centers,
                                                       const float* __restrict__ csq,
                                                       float* __restrict__ out) {
  __shared__ float s_a[TILE_M * LDA];  // 33792 B batch tile
  __shared__ float s_csq[256];
  __shared__ float s_xsq[TILE_M];
  __shared__ float s_rsum[TILE_M];

  const int tid = threadIdx.x;
  const int wave = tid >> 5;      // 0..7, each owns 32 output columns
  const int lane = tid & 31;
  const int lane_n = lane & 15;
  const int half = lane >> 4;
  const int row0 = blockIdx.x * TILE_M;
  const int col_base = wave * 32;

  if (tid < TILE_M) { s_xsq[tid] = 0.f; s_rsum[tid] = 0.f; }
  s_csq[tid] = csq[tid];
  __syncthreads();

  // Stage 64x128 batch tile into LDS; accumulate per-row ||x||^2 on the fly.
  {
    const int row = tid & 63;
    const int q = tid >> 6;  // 0..3 -> 32-float column chunk
    const float4* gsrc = (const float4*)(batch + (size_t)(row0 + row) * 128 + q * 32);
    float4* ldst = (float4*)(s_a + row * LDA + q * 32);
    float ss = 0.f;
#pragma unroll
    for (int i = 0; i < 8; ++i) {
      float4 v = gsrc[i];
      ldst[i] = v;
      ss += v.x * v.x + v.y * v.y + v.z * v.z + v.w * v.w;
    }
    atomicAdd(&s_xsq[row], ss);
  }
  __syncthreads();

  // cross = A(64x128) * centers^T slice (128 x 32 per wave), f32 WMMA 16x16x4.
  // A frag: lane (n, h) vgpr j -> A[M=n][K=kk+j+2h]   (float2 from LDS)
  // B frag: lane (n, h) vgpr j -> centers[n_col][kk+j+2h] (float2 from global, L2-resident)
  v8f acc[4][2] = {};
  const float* bp0 = centers + (size_t)(col_base + lane_n) * 128 + 2 * half;
  const float* bp1 = centers + (size_t)(col_base + 16 + lane_n) * 128 + 2 * half;
  const float* ap = s_a + lane_n * LDA + 2 * half;

#pragma unroll 4
  for (int kk = 0; kk < 128; kk += 4) {
    v2f b0 = *(const v2f*)(bp0 + kk);
    v2f b1 = *(const v2f*)(bp1 + kk);
#pragma unroll
    for (int rt = 0; rt < 4; ++rt) {
      v2f a = *(const v2f*)(ap + rt * (16 * LDA) + kk);
      acc[rt][0] = __builtin_amdgcn_wmma_f32_16x16x4_f32(false, a, false, b0, (short)0,
                                                         acc[rt][0], false, false);
      acc[rt][1] = __builtin_amdgcn_wmma_f32_16x16x4_f32(false, a, false, b1, (short)0,
                                                         acc[rt][1], false, false);
    }
  }

  // numerator = 1 / (1 + ||x||^2 + ||c||^2 - 2*cross); ALPHA=1 -> power==1.
  // C/D layout: vgpr j, lane (n,h) -> row = j + 8h (within 16-row tile), col = n.
#pragma unroll
  for (int rt = 0; rt < 4; ++rt) {
#pragma unroll
    for (int t = 0; t < 2; ++t) {
      float cs = s_csq[col_base + t * 16 + lane_n];
#pragma unroll
      for (int j = 0; j < 8; ++j) {
        int rowl = rt * 16 + j + 8 * half;
        float d2 = s_xsq[rowl] + cs - 2.f * acc[rt][t][j];
        acc[rt][t][j] = __builtin_amdgcn_rcpf(1.f + d2);
      }
    }
  }

  // Row sums over K=256: reduce t + 16 lanes via shfl_xor, one LDS atomic per
  // row per wave (8 waves each contribute their 32-column partial).
#pragma unroll
  for (int rt = 0; rt < 4; ++rt) {
#pragma unroll
    for (int j = 0; j < 8; ++j) {
      float s = acc[rt][0][j] + acc[rt][1][j];
      s += __shfl_xor(s, 1, 32);
      s += __shfl_xor(s, 2, 32);
      s += __shfl_xor(s, 4, 32);
      s += __shfl_xor(s, 8, 32);
      if (lane_n == 0) atomicAdd(&s_rsum[rt * 16 + j + 8 * half], s);
    }
  }
  __syncthreads();
  if (tid < TILE_M) s_rsum[tid] = __builtin_amdgcn_rcpf(s_rsum[tid]);
  __syncthreads();

  // Normalize and stream out (never re-read -> non-temporal stores).
  float* obase = out + (size_t)row0 * 256 + col_base + lane_n;
#pragma unroll
  for (int rt = 0; rt < 4; ++rt) {
#pragma unroll
    for (int t = 0; t < 2; ++t) {
#pragma unroll
      for (int j = 0; j < 8; ++j) {
        int rowl = rt * 16 + j + 8 * half;
        float v = acc[rt][t][j] * s_rsum[rowl];
        __builtin_nontemporal_store(v, obase + (size_t)rowl * 256 + t * 16);
      }
    }
  }
}

extern "C" void kernel_launch(void* const* d_in, const int* in_sizes, int n_in,
                              void* d_out, int out_size, void* d_ws, size_t ws_size,
                              hipStream_t stream) {
  const float* batch = (const float*)d_in[0];     // [B,128] f32
  const float* centers = (const float*)d_in[1];   // [256,128] f32
  float* out = (float*)d_out;                     // [B,256] f32
  float* csq = (float*)d_ws;                      // 256 floats scratch

  const int B = in_sizes[0] / 128;
  dec_csq_kernel<<<1, 256, 0, stream>>>(centers, csq);
  dec_main_kernel<<<B / TILE_M, 256, 0, stream>>>(batch, centers, csq, out);
}